// EnergyAttention_40673340293558
// MI455X (gfx1250) — compile-verified
//
#include <hip/hip_runtime.h>

// Problem constants (from the reference): B=256, C=512, D=1024, K=16.
#define B_N 256
#define C_N 512
#define D_N 1024
#define K_N 16

typedef float v2f __attribute__((ext_vector_type(2)));
typedef float v8f __attribute__((ext_vector_type(8)));

// Fused: energy GEMV (via V_WMMA_F32_16X16X4_F32) -> stable top-16 -> gather.
// One workgroup (1024 threads = 32 wave32) per batch element.
//
// B-fragment masking trick: s_w is laid out as [ W (1024 floats) | zeros (1024
// floats) ]. Lanes 0/16 (column N=0 of the B matrix) base their ds reads at
// s_w[0]; all other lanes base at s_w[D_N] (the zero mirror). The per-k B
// fragment is then a raw ds_load_b64 with NO per-iteration VALU masking;
// same-address reads in the zero region broadcast (no bank conflicts).
__launch_bounds__(1024, 1)
__global__ void energy_topk_gather_kernel(const float* __restrict__ inputs,
                                          const float* __restrict__ W,
                                          float* __restrict__ out) {
    __shared__ float s_w[2 * D_N];    // [W | zeros]  (8 KB)
    __shared__ float s_energy[C_N];   // per-clip energies (2 KB)
    __shared__ int   s_idx[K_N];      // top-16 clip indices

    const int b    = blockIdx.x;
    const int tid  = threadIdx.x;
    const int lane = tid & 31;        // wave32 lane
    const int wave = tid >> 5;        // 0..31

    // Stage W and the zero mirror into LDS (1024 threads, 2048 floats).
    s_w[tid]       = W[tid];
    s_w[D_N + tid] = 0.0f;
    __syncthreads();

    // ---------------- Phase 1: energies via WMMA f32 16x16x4 ----------------
    // Wave w computes 16 dot products for clips [16w, 16w+16).
    // A-frag (16x4, 32-bit): lanes 0-15 -> M=lane, K={0,1}; lanes 16-31 -> M=lane-16, K={2,3}.
    // B-frag (4x16): only column N=0 (lanes 0/16) carries W; other lanes read zeros.
    const int  c_base = wave * 16;
    const int  m      = lane & 15;
    const int  ksel   = (lane >> 4) << 1;              // 0 or 2
    const bool isN0   = (m == 0);                      // lanes 0 and 16
    const float* rowp = inputs + ((size_t)b * C_N + (c_base + m)) * D_N + ksel;
    const float* wptr = &s_w[(isN0 ? 0 : D_N) + ksel]; // live W or zero mirror

    v8f acc0 = {0.f, 0.f, 0.f, 0.f, 0.f, 0.f, 0.f, 0.f};
    v8f acc1 = {0.f, 0.f, 0.f, 0.f, 0.f, 0.f, 0.f, 0.f};

    #pragma unroll 8
    for (int k0 = 0; k0 < D_N; k0 += 8) {
        // Two independent WMMA accumulation chains (k0 and k0+4).
        v2f a0 = *(const v2f*)(rowp + k0);
        v2f a1 = *(const v2f*)(rowp + k0 + 4);
        v2f b0 = *(const v2f*)(wptr + k0);
        v2f b1 = *(const v2f*)(wptr + k0 + 4);
        acc0 = __builtin_amdgcn_wmma_f32_16x16x4_f32(false, a0, false, b0,
                                                     (short)0, acc0, false, false);
        acc1 = __builtin_amdgcn_wmma_f32_16x16x4_f32(false, a1, false, b1,
                                                     (short)0, acc1, false, false);
    }

    // Column N=0 of C/D lives in lane 0 (M=r in VGPR r) and lane 16 (M=8+r).
    if (isN0) {
        const int mbase = (lane >> 4) * 8;
        #pragma unroll
        for (int r = 0; r < 8; ++r)
            s_energy[c_base + mbase + r] = acc0[r] + acc1[r];
    }
    __syncthreads();

    // ---------------- Phase 2: stable top-16 by rank counting ----------------
    // rank(i) = #{ j : e_j > e_i  or (e_j == e_i and j < i) }  -> descending,
    // ties broken by lower index, exactly matching jax.lax.top_k ordering.
    // (softmax is monotonic, bias is a uniform shift: both order-preserving, skipped.)
    if (tid < C_N) {
        const float e = s_energy[tid];
        int rank = 0;
        #pragma unroll 8
        for (int j = 0; j < C_N; ++j) {
            const float ej = s_energy[j];
            rank += (ej > e) || (ej == e && j < tid);
        }
        if (rank < K_N) s_idx[rank] = tid;
    }
    __syncthreads();

    // ---------------- Phase 3: coalesced float4 gather ----------------
    // 1024 threads move 4 rows per pass (256 threads x float4 per row), 4 passes.
    const int rslot = tid >> 8;            // 0..3: which of 4 rows this pass
    const int col   = (tid & 255) * 4;     // float offset within row
    #pragma unroll
    for (int g = 0; g < 4; ++g) {
        const int kk  = g * 4 + rslot;
        const int src = s_idx[kk];
        const float4* ps = (const float4*)(inputs + ((size_t)b * C_N + src) * D_N + col);
        float4*       pd = (float4*)(out + ((size_t)b * K_N + kk) * D_N + col);
        *pd = *ps;
    }
}

extern "C" void kernel_launch(void* const* d_in, const int* in_sizes, int n_in,
                              void* d_out, int out_size, void* d_ws, size_t ws_size,
                              hipStream_t stream) {
    (void)in_sizes; (void)n_in; (void)d_ws; (void)ws_size; (void)out_size;
    const float* inputs = (const float*)d_in[0];  // [B, C, D] f32
    const float* W      = (const float*)d_in[1];  // [1, D]    f32
    // d_in[2] = bias (order-preserving shift, unobservable in output): unused.
    // d_in[3] = topk (fixed 16 per reference config): unused.
    float* out = (float*)d_out;                   // [B, K, D] f32

    energy_topk_gather_kernel<<<B_N, 1024, 0, stream>>>(inputs, W, out);
}